// EncModule_17360257810435
// MI455X (gfx1250) — compile-verified
//
#include <hip/hip_runtime.h>
#include <math.h>

typedef float v2f __attribute__((ext_vector_type(2)));
typedef float v8f __attribute__((ext_vector_type(8)));
typedef unsigned int u32x4 __attribute__((ext_vector_type(4)));
typedef int i32x8 __attribute__((ext_vector_type(8)));
typedef int i32x4 __attribute__((ext_vector_type(4)));

#define BATCH   8
#define CDIM    512
#define HW      9216
#define KCW     32
#define NCLS    19
#define NPIX    32
#define TPB     8            // pixel tiles per block (persistent E accumulation)
#define EPSB    1e-5f

// ---- workspace layout (floats) ----
#define WS_E      0            // BATCH*KCW*CDIM = 131072
#define WS_ASUM   131072       // BATCH*KCW     = 256
#define WS_GAMMA  131328       // BATCH*CDIM    = 4096
#define WS_ZERO_N 131328       // zero E + Asum

// ---- dynamic LDS layout (floats) for fused kernel ----
#define L_X    0               // 512*32 = 16384 (X tile, later h tile)
#define L_S2   16384           // 512
#define L_T2   16896           // 512
#define L_SL   17408           // 32*33 = 1056
#define L_A    18464           // 32*33 = 1056
#define L_X2   19520           // 32
#define L_C2   19552           // 32
#define L_RED  19584           // 256
#define L_TOT  19840           // floats -> 79360 bytes

// ---- Tensor Data Mover availability ----
#if defined(__has_builtin)
#if __has_builtin(__builtin_amdgcn_tensor_load_to_lds) && \
    __has_builtin(__builtin_amdgcn_s_wait_tensorcnt)
#define USE_TDM 1
#endif
#endif
#ifndef USE_TDM
#define USE_TDM 0
#endif

#if USE_TDM
// 2D tile DMA: global (tensor_d0 x tensor_d1, row stride stride0 elems, 4B elems)
// -> LDS contiguous tile (tile_d0 x tile_d1). Descriptor per CDNA5 ISA 8.3/8.4.
__device__ __forceinline__ void tdm_load_tile_f32(const void* gsrc, void* ldst,
                                                  unsigned tensor_d0, unsigned tensor_d1,
                                                  unsigned tile_d0, unsigned tile_d1,
                                                  unsigned stride0) {
  unsigned long long ga = (unsigned long long)gsrc;
  unsigned lds_off =
      (unsigned)(unsigned long long)(__attribute__((address_space(3))) void*)ldst;
  u32x4 g0;
  g0[0] = 1u;                                               // count=1, user D#
  g0[1] = lds_off;                                          // lds_addr
  g0[2] = (unsigned)(ga & 0xFFFFFFFFull);                   // global_addr[31:0]
  g0[3] = (unsigned)((ga >> 32) & 0x01FFFFFFull) | (2u << 30);  // addr[56:32] | type=2
  i32x8 g1;
  g1[0] = (int)(2u << 16);                                  // data_size=4B, mask=0
  g1[1] = (int)((tensor_d0 & 0xFFFFu) << 16);               // tensor_dim0[15:0]
  g1[2] = (int)((tensor_d0 >> 16) | ((tensor_d1 & 0xFFFFu) << 16));
  g1[3] = (int)((tensor_d1 >> 16) | (tile_d0 << 16));       // dim1 hi | tile_dim0
  g1[4] = (int)(tile_d1 & 0xFFFFu);                         // tile_dim1, tile_dim2=0
  g1[5] = (int)stride0;                                     // tensor_dim0_stride lo
  g1[6] = 0;                                                // stride0 hi | stride1 lo
  g1[7] = 0;                                                // stride1 hi
  i32x4 z4 = {};
#if __clang_major__ >= 23
  i32x8 z8 = {};
  __builtin_amdgcn_tensor_load_to_lds(g0, g1, z4, z4, z8, 0);
#else
  __builtin_amdgcn_tensor_load_to_lds(g0, g1, z4, z4, 0);
#endif
}
#endif

__global__ __launch_bounds__(256) void init_ws_kernel(float* __restrict__ ws) {
  int i = blockIdx.x * 256 + threadIdx.x;
  if (i < WS_ZERO_N) ws[i] = 0.f;
}

// Fused: 1x1 conv + BN2 + ReLU + soft-assign encoding over TPB pixel tiles,
// E fragments persistent in WMMA accumulators, one atomic round per block.
__global__ __launch_bounds__(256) void fused_enc_kernel(
    const float* __restrict__ x, const float* __restrict__ conv_w,
    const float* __restrict__ bn2_w, const float* __restrict__ bn2_b,
    const float* __restrict__ bn2_mean, const float* __restrict__ bn2_var,
    const float* __restrict__ codewords, const float* __restrict__ scale,
    float* __restrict__ Eacc, float* __restrict__ Asum) {
  extern __shared__ float smem[];
  float* lds_x  = smem + L_X;
  float* lds_s2 = smem + L_S2;
  float* lds_t2 = smem + L_T2;
  float* lds_sl = smem + L_SL;
  float* lds_A  = smem + L_A;
  float* lds_x2 = smem + L_X2;
  float* lds_c2 = smem + L_C2;
  float* red    = smem + L_RED;

  const int tid  = threadIdx.x;
  const int lane = tid & 31;
  const int wave = tid >> 5;
  const int b    = blockIdx.y;
  const int tile0 = blockIdx.x * TPB;
  const int l15  = lane & 15;
  const int hi   = lane >> 4;        // 0: lanes 0-15, 1: lanes 16-31
  const int khi  = hi * 2;           // K sub-offset for A/B frags

  // BN2 fold + codeword norms (loop-invariant)
  for (int c = tid; c < CDIM; c += 256) {
    float s = bn2_w[c] * rsqrtf(bn2_var[c] + EPSB);
    lds_s2[c] = s;
    lds_t2[c] = bn2_b[c] - bn2_mean[c] * s;
  }
  if (tid < KCW) {
    const float* cw = codewords + (size_t)tid * CDIM;
    float s = 0.f;
    for (int d = 0; d < CDIM; ++d) { float v = cw[d]; s += v * v; }
    lds_c2[tid] = s;
  }

  // Persistent E accumulators: wave owns k-tile (wave&1), channel group (wave>>1)
  const int emt = wave & 1;
  const int edg = wave >> 1;
  v8f eacc[8];
#pragma unroll
  for (int j = 0; j < 8; ++j) eacc[j] = {};
  float asum_reg = 0.f;

  for (int t = 0; t < TPB; ++t) {
    const int n0 = (tile0 + t) * NPIX;
    __syncthreads();  // previous iteration's readers of lds_x / lds_A are done

    // ---- stage X tile (CDIM x NPIX) into LDS ----
#if USE_TDM
    if (wave == 0) {
      const float* gsrc = x + (size_t)b * CDIM * HW + n0;
      tdm_load_tile_f32(gsrc, lds_x, HW, CDIM, NPIX, CDIM, HW);
      __builtin_amdgcn_s_wait_tensorcnt(0);
    }
#else
    {
      const float* xb = x + (size_t)b * CDIM * HW + n0;
      for (int i = tid; i < CDIM * NPIX; i += 256) {
        int c = i >> 5, p = i & 31;
        lds_x[i] = xb[(size_t)c * HW + p];
      }
    }
#endif
    __syncthreads();

    // ---- GEMM: h(512x32) = conv_w(512x512) * X(512x32); wave owns 64 rows ----
    const int o0 = wave * 64;
    v8f acc[4][2];
#pragma unroll
    for (int mt = 0; mt < 4; ++mt)
#pragma unroll
      for (int nt = 0; nt < 2; ++nt) acc[mt][nt] = {};

    for (int kk = 0; kk < CDIM; kk += 4) {
      v2f bfrag[2];
#pragma unroll
      for (int nt = 0; nt < 2; ++nt) {
        int col = nt * 16 + l15;
        bfrag[nt][0] = lds_x[(kk + khi) * NPIX + col];
        bfrag[nt][1] = lds_x[(kk + khi + 1) * NPIX + col];
      }
#pragma unroll
      for (int mt = 0; mt < 4; ++mt) {
        const float* wr = conv_w + (size_t)(o0 + mt * 16 + l15) * CDIM + kk + khi;
        v2f afrag;
        afrag[0] = wr[0];
        afrag[1] = wr[1];
#pragma unroll
        for (int nt = 0; nt < 2; ++nt)
          acc[mt][nt] = __builtin_amdgcn_wmma_f32_16x16x4_f32(
              false, afrag, false, bfrag[nt], (short)0, acc[mt][nt], false, false);
      }
    }
    __syncthreads();  // everyone done reading X

    // BN2 + ReLU, write h into lds_x (overwrite)
#pragma unroll
    for (int mt = 0; mt < 4; ++mt)
#pragma unroll
      for (int nt = 0; nt < 2; ++nt)
#pragma unroll
        for (int r = 0; r < 8; ++r) {
          int o = o0 + mt * 16 + r + hi * 8;  // C/D layout: M = r + 8*hi
          int p = nt * 16 + l15;
          float v = fmaxf(acc[mt][nt][r] * lds_s2[o] + lds_t2[o], 0.f);
          lds_x[o * NPIX + p] = v;
        }
    __syncthreads();

    // ---- x2[p] = sum_c h[c][p]^2 ----
    {
      int p = tid & 31, seg = tid >> 5;
      float s = 0.f;
      for (int c = seg * 64; c < seg * 64 + 64; ++c) {
        float v = lds_x[c * NPIX + p];
        s += v * v;
      }
      red[tid] = s;
    }
    __syncthreads();
    if (tid < NPIX) {
      float s = 0.f;
#pragma unroll
      for (int seg = 0; seg < 8; ++seg) s += red[seg * 32 + tid];
      lds_x2[tid] = s;
    }
    __syncthreads();

    // ---- xc(32x32) = codewords(32x512) * h(512x32), waves 0..3 ----
    if (wave < 4) {
      int mt = wave >> 1, nt = wave & 1;
      v8f xcacc = {};
      for (int cc = 0; cc < CDIM; cc += 4) {
        const float* cw = codewords + (size_t)(mt * 16 + l15) * CDIM + cc + khi;
        v2f a;  a[0] = cw[0]; a[1] = cw[1];
        int col = nt * 16 + l15;
        v2f bb; bb[0] = lds_x[(cc + khi) * NPIX + col];
                bb[1] = lds_x[(cc + khi + 1) * NPIX + col];
        xcacc = __builtin_amdgcn_wmma_f32_16x16x4_f32(
            false, a, false, bb, (short)0, xcacc, false, false);
      }
#pragma unroll
      for (int r = 0; r < 8; ++r) {
        int k = mt * 16 + r + hi * 8;
        int p = nt * 16 + l15;
        float v = scale[k] * (lds_x2[p] - 2.f * xcacc[r] + lds_c2[k]);
        lds_sl[p * 33 + k] = v;  // padded stride 33 (bank-conflict free)
      }
    }
    __syncthreads();

    // ---- softmax over k per pixel ----
    if (tid < NPIX) {
      int p = tid;
      float m = -1e30f;
      for (int k = 0; k < KCW; ++k) m = fmaxf(m, lds_sl[p * 33 + k]);
      float s = 0.f;
      for (int k = 0; k < KCW; ++k) {
        float e = __expf(lds_sl[p * 33 + k] - m);
        lds_A[p * 33 + k] = e;
        s += e;
      }
      float inv = 1.f / s;
      for (int k = 0; k < KCW; ++k) lds_A[p * 33 + k] *= inv;
    }
    __syncthreads();

    // per-codeword assignment mass, accumulated in register
    if (tid < KCW) {
      float s = 0.f;
      for (int p = 0; p < NPIX; ++p) s += lds_A[p * 33 + tid];
      asum_reg += s;
    }

    // ---- E partial: (32 x NPIX) * (NPIX x 512), accumulate in registers ----
#pragma unroll
    for (int j = 0; j < 8; ++j) {
      int d0 = (edg * 8 + j) * 16;
#pragma unroll
      for (int pp = 0; pp < NPIX; pp += 4) {
        v2f a;
        a[0] = lds_A[(pp + khi) * 33 + emt * 16 + l15];
        a[1] = lds_A[(pp + khi + 1) * 33 + emt * 16 + l15];
        v2f bb;
        bb[0] = lds_x[(d0 + l15) * NPIX + pp + khi];
        bb[1] = lds_x[(d0 + l15) * NPIX + pp + khi + 1];
        eacc[j] = __builtin_amdgcn_wmma_f32_16x16x4_f32(
            false, a, false, bb, (short)0, eacc[j], false, false);
      }
    }
  }

  // ---- one atomic round per block into L2-resident E ----
#pragma unroll
  for (int j = 0; j < 8; ++j) {
    int d0 = (edg * 8 + j) * 16;
#pragma unroll
    for (int r = 0; r < 8; ++r) {
      int k = emt * 16 + r + hi * 8;
      int d = d0 + l15;
      unsafeAtomicAdd(&Eacc[((size_t)b * KCW + k) * CDIM + d], eacc[j][r]);
    }
  }
  if (tid < KCW) unsafeAtomicAdd(&Asum[b * KCW + tid], asum_reg);
}

// Finalize: E -> en -> gamma (sigmoid gate) and se logits. One block per batch.
__global__ __launch_bounds__(256) void finalize_kernel(
    const float* __restrict__ Eacc, const float* __restrict__ Asum,
    const float* __restrict__ codewords,
    const float* __restrict__ bn1_w, const float* __restrict__ bn1_b,
    const float* __restrict__ bn1_mean, const float* __restrict__ bn1_var,
    const float* __restrict__ fc_w, const float* __restrict__ fc_b,
    const float* __restrict__ se_w, const float* __restrict__ se_b,
    float* __restrict__ gamma, float* __restrict__ se_out) {
  __shared__ float s_en[CDIM];
  __shared__ float s1[KCW], t1[KCW], s_as[KCW];
  const int b = blockIdx.x, tid = threadIdx.x;
  if (tid < KCW) {
    float s = bn1_w[tid] * rsqrtf(bn1_var[tid] + EPSB);
    s1[tid] = s;
    t1[tid] = bn1_b[tid] - bn1_mean[tid] * s;
    s_as[tid] = Asum[b * KCW + tid];
  }
  __syncthreads();
  for (int c = tid; c < CDIM; c += 256) {
    float a = 0.f;
    for (int k = 0; k < KCW; ++k) {
      float e = Eacc[((size_t)b * KCW + k) * CDIM + c] -
                s_as[k] * codewords[(size_t)k * CDIM + c];
      a += fmaxf(e * s1[k] + t1[k], 0.f);
    }
    s_en[c] = a * (1.f / KCW);
  }
  __syncthreads();
  for (int c = tid; c < CDIM; c += 256) {
    const float* w = fc_w + (size_t)c * CDIM;
    float g = fc_b[c];
    for (int d = 0; d < CDIM; ++d) g += s_en[d] * w[d];
    gamma[b * CDIM + c] = 1.f / (1.f + __expf(-g));
  }
  if (tid < NCLS) {
    const float* w = se_w + (size_t)tid * CDIM;
    float s = se_b[tid];
    for (int d = 0; d < CDIM; ++d) s += s_en[d] * w[d];
    se_out[b * NCLS + tid] = s;
  }
}

// out = relu(x * (1 + gamma)), float4-vectorized (HW divisible by 4)
__global__ __launch_bounds__(256) void gate_kernel(
    const float* __restrict__ x, const float* __restrict__ gamma,
    float* __restrict__ out) {
  size_t i4 = (size_t)blockIdx.x * 256 + threadIdx.x;
  size_t base = i4 * 4;
  int bc = (int)(base / HW);  // == b*CDIM + c
  float g = 1.f + gamma[bc];
  float4 xv = ((const float4*)x)[i4];
  float4 o;
  o.x = fmaxf(xv.x * g, 0.f);
  o.y = fmaxf(xv.y * g, 0.f);
  o.z = fmaxf(xv.z * g, 0.f);
  o.w = fmaxf(xv.w * g, 0.f);
  ((float4*)out)[i4] = o;
}

extern "C" void kernel_launch(void* const* d_in, const int* in_sizes, int n_in,
                              void* d_out, int out_size, void* d_ws, size_t ws_size,
                              hipStream_t stream) {
  const float* x        = (const float*)d_in[0];
  const float* conv_w   = (const float*)d_in[1];
  const float* bn2_w    = (const float*)d_in[2];
  const float* bn2_b    = (const float*)d_in[3];
  const float* bn2_mean = (const float*)d_in[4];
  const float* bn2_var  = (const float*)d_in[5];
  const float* codewords= (const float*)d_in[6];
  const float* scale    = (const float*)d_in[7];
  const float* bn1_w    = (const float*)d_in[8];
  const float* bn1_b    = (const float*)d_in[9];
  const float* bn1_mean = (const float*)d_in[10];
  const float* bn1_var  = (const float*)d_in[11];
  const float* fc_w     = (const float*)d_in[12];
  const float* fc_b     = (const float*)d_in[13];
  const float* se_w     = (const float*)d_in[14];
  const float* se_b     = (const float*)d_in[15];

  float* out    = (float*)d_out;
  float* se_out = out + (size_t)BATCH * CDIM * HW;  // tuple output: (out, se)
  float* ws     = (float*)d_ws;
  float* Eacc   = ws + WS_E;
  float* Asum   = ws + WS_ASUM;
  float* gamma  = ws + WS_GAMMA;

  init_ws_kernel<<<(WS_ZERO_N + 255) / 256, 256, 0, stream>>>(ws);

  dim3 g1(HW / (NPIX * TPB), BATCH);  // (36, 8)
  fused_enc_kernel<<<g1, 256, L_TOT * sizeof(float), stream>>>(
      x, conv_w, bn2_w, bn2_b, bn2_mean, bn2_var, codewords, scale, Eacc, Asum);

  finalize_kernel<<<BATCH, 256, 0, stream>>>(
      Eacc, Asum, codewords, bn1_w, bn1_b, bn1_mean, bn1_var,
      fc_w, fc_b, se_w, se_b, gamma, se_out);

  size_t n4 = (size_t)BATCH * CDIM * HW / 4;  // 9437184
  gate_kernel<<<(unsigned)(n4 / 256), 256, 0, stream>>>(x, gamma, out);
}